// DenseGrid_50328426775010
// MI455X (gfx1250) — compile-verified
//
#include <hip/hip_runtime.h>

// ---------------------------------------------------------------------------
// Multi-LOD trilinear interpolation (dense feature grids), MI455X / gfx1250.
// Gather-bandwidth bound: LOD0 staged in LDS via async-to-LDS (CDNA5 path),
// LODs 1-4 gathered from global with 64B-contiguous corner-pair bursts,
// NT hints on the streaming point/output traffic to protect L2 for grids.
// ---------------------------------------------------------------------------

#define FEAT 8
#define CB0_RES 16
#define CB0_FLOATS (CB0_RES * CB0_RES * CB0_RES * FEAT)   // 32768 floats = 128 KB
#define CB0_VEC4   (CB0_FLOATS / 4)                       // 8192 vec4

typedef float v4f __attribute__((ext_vector_type(4)));
typedef int   v4i __attribute__((ext_vector_type(4)));

#if defined(__gfx1250__) && __has_builtin(__builtin_amdgcn_global_load_async_to_lds_b128)
#define HAVE_ASYNC_LDS 1
typedef __attribute__((address_space(1))) v4i as1_v4i;
typedef __attribute__((address_space(3))) v4i as3_v4i;
#endif

__device__ __forceinline__ void fma4(v4f& a, float w, const v4f& v) {
    a.x = fmaf(w, v.x, a.x);
    a.y = fmaf(w, v.y, a.y);
    a.z = fmaf(w, v.z, a.z);
    a.w = fmaf(w, v.w, a.w);
}

// Accumulate one LOD. g points at the grid as v4f[res^3 * 2].
// Works for both global memory and LDS (inlined; addrspace inferred).
template <int RES>
__device__ __forceinline__ void lod_accum(const v4f* __restrict__ g,
                                          float px, float py, float pz,
                                          v4f& a0, v4f& a1) {
    constexpr float S    = (float)(RES - 1);
    constexpr float MX   = (float)(RES - 2);
    constexpr int   RES2 = RES * RES;

    const float xs = px * S, ys = py * S, zs = pz * S;
    float fx = floorf(xs), fy = floorf(ys), fz = floorf(zs);
    fx = fminf(fmaxf(fx, 0.0f), MX);
    fy = fminf(fmaxf(fy, 0.0f), MX);
    fz = fminf(fmaxf(fz, 0.0f), MX);
    const float tx = xs - fx, ty = ys - fy, tz = zs - fz;
    const int ix = (int)fx, iy = (int)fy, iz = (int)fz;

    const float wx0 = 1.0f - tx, wx1 = tx;
    const float wy0 = 1.0f - ty, wy1 = ty;
    const float wz0 = 1.0f - tz, wz1 = tz;

    const int row00 = ix + iy * RES + iz * RES2;

    // Each (j,k) pair: corners i=0,1 are 64B contiguous -> one cacheline burst.
#define PAIR(ROW, WYZ)                                                    \
    {                                                                     \
        const size_t b = (size_t)(ROW) * 2u;                              \
        const v4f v0 = g[b + 0], v1 = g[b + 1];                           \
        const v4f v2 = g[b + 2], v3 = g[b + 3];                           \
        const float w0 = (WYZ) * wx0, w1 = (WYZ) * wx1;                   \
        fma4(a0, w0, v0); fma4(a1, w0, v1);                               \
        fma4(a0, w1, v2); fma4(a1, w1, v3);                               \
    }

    PAIR(row00,               wy0 * wz0)
    PAIR(row00 + RES,         wy1 * wz0)
    PAIR(row00 + RES2,        wy0 * wz1)
    PAIR(row00 + RES + RES2,  wy1 * wz1)
#undef PAIR
}

__global__ void __launch_bounds__(256)
grid_trilerp_kernel(const float* __restrict__ pts,
                    const float* __restrict__ cb0,
                    const float* __restrict__ cb1,
                    const float* __restrict__ cb2,
                    const float* __restrict__ cb3,
                    const float* __restrict__ cb4,
                    float* __restrict__ out, int n) {
    // LOD0 grid staged in LDS: 128 KB of the 320 KB per WGP.
    __shared__ float s_cb0[CB0_FLOATS];

    {
        v4f* dst4 = (v4f*)s_cb0;
        v4f* src4 = (v4f*)cb0;
#if defined(HAVE_ASYNC_LDS)
        for (int t = threadIdx.x; t < CB0_VEC4; t += blockDim.x) {
            __builtin_amdgcn_global_load_async_to_lds_b128(
                (as1_v4i*)(src4 + t), (as3_v4i*)(dst4 + t), 0, 0);
        }
#if __has_builtin(__builtin_amdgcn_s_wait_asynccnt)
        __builtin_amdgcn_s_wait_asynccnt(0);
#else
        asm volatile("s_wait_asynccnt 0" ::: "memory");
#endif
#else
        for (int t = threadIdx.x; t < CB0_VEC4; t += blockDim.x) {
            dst4[t] = src4[t];
        }
#endif
        __syncthreads();
    }

    const v4f* g1 = (const v4f*)cb1;
    const v4f* g2 = (const v4f*)cb2;
    const v4f* g3 = (const v4f*)cb3;
    const v4f* g4 = (const v4f*)cb4;
    const v4f* g0 = (const v4f*)s_cb0;
    v4f*       o4 = (v4f*)out;

    const int stride = (int)(gridDim.x * blockDim.x);
    for (int i = (int)(blockIdx.x * blockDim.x + threadIdx.x); i < n; i += stride) {
        // Streaming reads: non-temporal so they don't evict grid data from L2.
        const float px = __builtin_nontemporal_load(pts + 3 * (size_t)i + 0);
        const float py = __builtin_nontemporal_load(pts + 3 * (size_t)i + 1);
        const float pz = __builtin_nontemporal_load(pts + 3 * (size_t)i + 2);

        v4f a0 = {0.f, 0.f, 0.f, 0.f};
        v4f a1 = {0.f, 0.f, 0.f, 0.f};

        lod_accum<16>(g0, px, py, pz, a0, a1);    // from LDS (ds pipe)
        lod_accum<32>(g1, px, py, pz, a0, a1);    // L2-resident
        lod_accum<64>(g2, px, py, pz, a0, a1);    // L2-resident
        lod_accum<128>(g3, px, py, pz, a0, a1);   // L2-resident (64 MB)
        lod_accum<256>(g4, px, py, pz, a0, a1);   // 512 MB, partially L2

        // Streaming writes: non-temporal, two b128 stores (32 B/point).
        __builtin_nontemporal_store(a0, o4 + 2 * (size_t)i + 0);
        __builtin_nontemporal_store(a1, o4 + 2 * (size_t)i + 1);
    }
}

extern "C" void kernel_launch(void* const* d_in, const int* in_sizes, int n_in,
                              void* d_out, int out_size, void* d_ws, size_t ws_size,
                              hipStream_t stream) {
    const float* pts = (const float*)d_in[0];
    const float* cb0 = (const float*)d_in[1];
    const float* cb1 = (const float*)d_in[2];
    const float* cb2 = (const float*)d_in[3];
    const float* cb3 = (const float*)d_in[4];
    const float* cb4 = (const float*)d_in[5];
    float*       out = (float*)d_out;

    const int n = in_sizes[0] / 3;   // 2,000,000 points

    // 256 threads = 8 wave32 per block; 128 KB LDS -> 2 blocks per WGP.
    // Grid-stride over 1024 blocks amortizes the LDS staging (~128 KB per
    // block vs ~2.5 MB of gather work per block).
    const int block = 256;
    const int grid  = 1024;
    grid_trilerp_kernel<<<grid, block, 0, stream>>>(pts, cb0, cb1, cb2, cb3, cb4,
                                                    out, n);
}